// SAGE_77610059039112
// MI455X (gfx1250) — compile-verified
//
#include <hip/hip_runtime.h>
#include <hip/hip_bf16.h>

typedef __attribute__((ext_vector_type(2))) float v2f;
typedef __attribute__((ext_vector_type(8))) float v8f;

// ---------------------------------------------------------------------------
// zero-init (float4 vectorized)
// ---------------------------------------------------------------------------
__global__ void zero_kernel(float4* __restrict__ p, long n4) {
  long i = (long)blockIdx.x * blockDim.x + threadIdx.x;
  if (i < n4) p[i] = make_float4(0.f, 0.f, 0.f, 0.f);
}

// ---------------------------------------------------------------------------
// WMMA fp32 GEMM: OUT[M,N] = act( A1[M,K]@W1[K,N] (+ A2[M,K]@W2[K,N]) + bias )
// One wave32 per 16x16 output tile, V_WMMA_F32_16X16X4_F32, K fully unrolled.
// ACT: 0 = none, 1 = relu, 2 = tanh
// ---------------------------------------------------------------------------
template<int ACT, bool DUAL, int K, int N>
__global__ void sage_gemm_wmma(const float* __restrict__ A1, const float* __restrict__ W1,
                               const float* __restrict__ A2, const float* __restrict__ W2,
                               const float* __restrict__ bias, float* __restrict__ OUT,
                               int M) {
  const int lane = threadIdx.x & 31;
  const int half = lane >> 4;          // 0 -> K pair (0,1), 1 -> K pair (2,3)
  const int r    = lane & 15;
  const int wavesPerBlock = blockDim.x >> 5;
  const int tile = blockIdx.x * wavesPerBlock + (threadIdx.x >> 5);
  constexpr int nTilesN = N >> 4;
  const int tileM = tile / nTilesN;
  const int tileN = tile - tileM * nTilesN;
  const int row0 = tileM << 4;
  if (row0 >= M) return;
  const int col0 = tileN << 4;

  v8f c = {0.f, 0.f, 0.f, 0.f, 0.f, 0.f, 0.f, 0.f};

  // --- accumulate A1 @ W1 ---
  {
    const float* ap = A1 + (size_t)(row0 + r) * K + 2 * half;
    const float* wp = W1 + (size_t)(2 * half) * N + col0 + r;
#pragma unroll
    for (int k0 = 0; k0 < K; k0 += 4) {
      float2 av = *(const float2*)(ap + k0);
      v2f a; a.x = av.x; a.y = av.y;
      v2f b; b.x = wp[(size_t)k0 * N]; b.y = wp[(size_t)k0 * N + N];
      c = __builtin_amdgcn_wmma_f32_16x16x4_f32(false, a, false, b, (short)0, c, false, false);
    }
  }
  // --- accumulate A2 @ W2 into same tile ---
  if constexpr (DUAL) {
    const float* ap = A2 + (size_t)(row0 + r) * K + 2 * half;
    const float* wp = W2 + (size_t)(2 * half) * N + col0 + r;
#pragma unroll
    for (int k0 = 0; k0 < K; k0 += 4) {
      float2 av = *(const float2*)(ap + k0);
      v2f a; a.x = av.x; a.y = av.y;
      v2f b; b.x = wp[(size_t)k0 * N]; b.y = wp[(size_t)k0 * N + N];
      c = __builtin_amdgcn_wmma_f32_16x16x4_f32(false, a, false, b, (short)0, c, false, false);
    }
  }

  // C/D layout: VGPR i -> row (i + 8*half), col (lane & 15) within tile
  const float bv = bias[col0 + r];
#pragma unroll
  for (int i = 0; i < 8; ++i) {
    float v = c[i] + bv;
    if constexpr (ACT == 1) v = fmaxf(v, 0.f);
    if constexpr (ACT == 2) v = tanhf(v);
    OUT[(size_t)(row0 + i + 8 * half) * N + col0 + r] = v;
  }
}

// ---------------------------------------------------------------------------
// Edge scatter-max, F=128: one wave per edge; lane loads float4 (b128) of the
// pooled source row, then 4x GLOBAL_ATOMIC_MAX_U32 into the dst row.
// Valid because relu => all values >= 0 (uint order == IEEE order), and the
// zero-initialized agg implements the empty-neighborhood -> 0 convention.
// ---------------------------------------------------------------------------
__global__ void edge_max_f128(const float* __restrict__ m, const int* __restrict__ src,
                              const int* __restrict__ dst, unsigned int* __restrict__ agg,
                              int nE) {
  int e    = blockIdx.x * (blockDim.x >> 5) + (threadIdx.x >> 5);
  int lane = threadIdx.x & 31;
  if (e >= nE) return;
  size_t s = (size_t)src[e] * 128;
  size_t d = (size_t)dst[e] * 128;
  float4 v = *(const float4*)(m + s + lane * 4);
  unsigned int* a = agg + d + lane * 4;
  atomicMax(a + 0, __float_as_uint(v.x));
  atomicMax(a + 1, __float_as_uint(v.y));
  atomicMax(a + 2, __float_as_uint(v.z));
  atomicMax(a + 3, __float_as_uint(v.w));
}

// F=64: two edges per wave (16 lanes x float4 per edge)
__global__ void edge_max_f64(const float* __restrict__ m, const int* __restrict__ src,
                             const int* __restrict__ dst, unsigned int* __restrict__ agg,
                             int nE) {
  int wave = blockIdx.x * (blockDim.x >> 5) + (threadIdx.x >> 5);
  int lane = threadIdx.x & 31;
  int e = wave * 2 + (lane >> 4);
  if (e >= nE) return;
  int j = lane & 15;
  size_t s = (size_t)src[e] * 64;
  size_t d = (size_t)dst[e] * 64;
  float4 v = *(const float4*)(m + s + j * 4);
  unsigned int* a = agg + d + j * 4;
  atomicMax(a + 0, __float_as_uint(v.x));
  atomicMax(a + 1, __float_as_uint(v.y));
  atomicMax(a + 2, __float_as_uint(v.z));
  atomicMax(a + 3, __float_as_uint(v.w));
}

// ---------------------------------------------------------------------------
extern "C" void kernel_launch(void* const* d_in, const int* in_sizes, int n_in,
                              void* d_out, int out_size, void* d_ws, size_t ws_size,
                              hipStream_t stream) {
  const float* x   = (const float*)d_in[0];
  const int*   src = (const int*)d_in[1];
  const int*   dst = (const int*)d_in[2];
  const float* Wp1 = (const float*)d_in[3];
  const float* bp1 = (const float*)d_in[4];
  const float* Ws1 = (const float*)d_in[5];
  const float* Wn1 = (const float*)d_in[6];
  const float* b1  = (const float*)d_in[7];
  const float* Wp3 = (const float*)d_in[8];
  const float* bp3 = (const float*)d_in[9];
  const float* Ws3 = (const float*)d_in[10];
  const float* Wn3 = (const float*)d_in[11];
  const float* b3  = (const float*)d_in[12];
  float* out = (float*)d_out;

  const int IN_F = 128;
  const int HID_F = 64;
  const int M  = in_sizes[0] / IN_F;   // 100000 nodes
  const int nE = in_sizes[1];          // 1600000 edges

  // workspace: m [M,128] | agg [M,128] | h [M,64]  (~128 MB total)
  float* ws_m   = (float*)d_ws;
  float* ws_agg = ws_m + (size_t)M * IN_F;
  float* ws_h   = ws_agg + (size_t)M * IN_F;

  const int TPB = 256;     // 8 wave32s per block
  const int WPB = 8;
  auto cdiv = [](long a, long b) { return (int)((a + b - 1) / b); };
  const int mTiles = cdiv(M, 16);

  // ---- Layer 1 ----
  // 1) m1 = relu(x @ Wp1 + bp1)   [M,128]
  sage_gemm_wmma<1, false, 128, 128><<<cdiv((long)mTiles * 8, WPB), TPB, 0, stream>>>(
      x, Wp1, nullptr, nullptr, bp1, ws_m, M);
  // 2) agg1 = 0
  {
    long n4 = (long)M * IN_F / 4;
    zero_kernel<<<cdiv(n4, TPB), TPB, 0, stream>>>((float4*)ws_agg, n4);
  }
  // 3) agg1[v] = max over edges of m1[src]   (L2-resident scatter-max)
  edge_max_f128<<<cdiv(nE, WPB), TPB, 0, stream>>>(ws_m, src, dst, (unsigned int*)ws_agg, nE);
  // 4) h = tanh(x @ Ws1 + agg1 @ Wn1 + b1)   [M,64]
  sage_gemm_wmma<2, true, 128, 64><<<cdiv((long)mTiles * 4, WPB), TPB, 0, stream>>>(
      x, Ws1, ws_agg, Wn1, b1, ws_h, M);

  // ---- Layer 2 ----
  // 5) m3 = relu(h @ Wp3 + bp3)   [M,64]  (reuse ws_m)
  sage_gemm_wmma<1, false, 64, 64><<<cdiv((long)mTiles * 4, WPB), TPB, 0, stream>>>(
      ws_h, Wp3, nullptr, nullptr, bp3, ws_m, M);
  // 6) agg3 = 0
  {
    long n4 = (long)M * HID_F / 4;
    zero_kernel<<<cdiv(n4, TPB), TPB, 0, stream>>>((float4*)ws_agg, n4);
  }
  // 7) scatter-max at F=64 (2 edges per wave)
  edge_max_f64<<<cdiv(nE, (long)WPB * 2), TPB, 0, stream>>>(ws_m, src, dst,
                                                            (unsigned int*)ws_agg, nE);
  // 8) out = h @ Ws3 + agg3 @ Wn3 + b3   [M,64]
  sage_gemm_wmma<0, true, 64, 64><<<cdiv((long)mTiles * 4, WPB), TPB, 0, stream>>>(
      ws_h, Ws3, ws_agg, Wn3, b3, out, M);
}